// FPNRPNBoxSelector_75866302316589
// MI455X (gfx1250) — compile-verified
//
#include <hip/hip_runtime.h>
#include <math.h>

#ifndef __has_builtin
#define __has_builtin(x) 0
#endif

#if __has_builtin(__builtin_amdgcn_global_load_async_to_lds_b128) && __has_builtin(__builtin_amdgcn_s_wait_asynccnt)
#define USE_ASYNC_LDS 1
#else
#define USE_ASYNC_LDS 0
#endif

#define NIMG 8
#define NLVL 3
#define NA 3
#define KSEL 1000
#define KPAD 1024
#define FINPAD 4096
#define NEG_INF_F (-1e30f)
#define NMS_TH 0.7f
#define IMW 1024.0f
#define IMH 1024.0f
#define XCLIP 4.135166556742356f   /* ln(1000/16) */

// Monotone float -> uint key (larger float => larger key)
__device__ __forceinline__ unsigned f2key(float f) {
  unsigned u = __float_as_uint(f);
  return (u & 0x80000000u) ? ~u : (u | 0x80000000u);
}

// ---------------------------------------------------------------------------
// Kernel 1: per (img,level) radix-select top-1000 logits, sort descending,
// decode + clip boxes. One workgroup (32 waves) per (img,level).
// ---------------------------------------------------------------------------
extern "C" __global__ __launch_bounds__(1024)
void k_topk_decode(const float* __restrict__ obj0, const float* __restrict__ obj1, const float* __restrict__ obj2,
                   const float* __restrict__ brg0, const float* __restrict__ brg1, const float* __restrict__ brg2,
                   const float* __restrict__ anc0, const float* __restrict__ anc1, const float* __restrict__ anc2,
                   float* __restrict__ sel_score, float* __restrict__ sel_box, int* __restrict__ sel_idx)
{
  const int blk = blockIdx.x;
  const int img = blk / NLVL, lvl = blk - img * NLVL;
  const int H = 256 >> lvl, W = H, HW = H * W, M = HW * NA;
  const float* obj = (lvl == 0) ? obj0 : ((lvl == 1) ? obj1 : obj2);
  const float* brg = (lvl == 0) ? brg0 : ((lvl == 1) ? brg1 : brg2);
  const float* anc = (lvl == 0) ? anc0 : ((lvl == 1) ? anc1 : anc2);
  const float* objn = obj + (size_t)img * M;                 // (3,H,W) for this image
  const float* brgn = brg + (size_t)img * (NA * 4) * HW;     // (12,H,W)
  const float* ancn = anc + (size_t)img * M * 4;             // (M,4)
  const int t = threadIdx.x, NT = blockDim.x;

  __shared__ int   hist[4096];
  __shared__ float ls[KPAD];
  __shared__ int   li[KPAD];
  __shared__ int   sc[8];   // 0:cntA 1:cntB 2:T1 3:T2 4:above1 5:definite

  for (int b = t; b < 4096; b += NT) hist[b] = 0;
  if (t < 8) sc[t] = 0;
  __syncthreads();

  // Pass 1: histogram of top 12 key bits (streaming scan, prefetch ahead).
  #pragma unroll 1
  for (int j = t; j < M; j += NT) {
    __builtin_prefetch(objn + j + 8192, 0, 0);
    atomicAdd(&hist[f2key(objn[j]) >> 20], 1);
  }
  __syncthreads();
  if (t == 0) {
    int cum = 0, Tb = 0;
    #pragma unroll 1
    for (int b = 4095; b >= 0; b--) {
      if (cum + hist[b] >= KSEL) { Tb = b; break; }
      cum += hist[b];
    }
    sc[2] = Tb; sc[4] = cum;
  }
  __syncthreads();
  const int T1 = sc[2];
  for (int b = t; b < 4096; b += NT) hist[b] = 0;
  __syncthreads();

  // Pass 2: refine boundary bin on next 12 key bits.
  #pragma unroll 1
  for (int j = t; j < M; j += NT) {
    unsigned k = f2key(objn[j]);
    if ((int)(k >> 20) == T1) atomicAdd(&hist[(k >> 8) & 0xFFFu], 1);
  }
  __syncthreads();
  if (t == 0) {
    int need = KSEL - sc[4];
    int cum = 0, Tb = 0;
    #pragma unroll 1
    for (int b = 4095; b >= 0; b--) {
      if (cum + hist[b] >= need) { Tb = b; break; }
      cum += hist[b];
    }
    sc[3] = Tb; sc[5] = sc[4] + cum;
  }
  __syncthreads();
  const int T2 = sc[3], DEF = sc[5];
  ls[t] = NEG_INF_F; li[t] = -1;
  __syncthreads();

  // Compaction A: strictly above the 24-bit threshold prefix (DEF < KSEL elements).
  #pragma unroll 1
  for (int j = t; j < M; j += NT) {
    float v = objn[j];
    unsigned k = f2key(v);
    int b1 = (int)(k >> 20), b2 = (int)((k >> 8) & 0xFFFu);
    if (b1 > T1 || (b1 == T1 && b2 > T2)) {
      int slot = atomicAdd(&sc[0], 1);
      int a = j / HW, hw = j - a * HW;
      ls[slot] = v; li[slot] = hw * NA + a;   // reference flat index (H,W,A order)
    }
  }
  __syncthreads();
  // Compaction B: boundary prefix fills remaining slots up to KSEL.
  #pragma unroll 1
  for (int j = t; j < M; j += NT) {
    float v = objn[j];
    unsigned k = f2key(v);
    if ((int)(k >> 20) == T1 && (int)((k >> 8) & 0xFFFu) == T2) {
      int slot = DEF + atomicAdd(&sc[1], 1);
      if (slot < KSEL) {
        int a = j / HW, hw = j - a * HW;
        ls[slot] = v; li[slot] = hw * NA + a;
      }
    }
  }

  // Bitonic sort of 1024 entries, descending (sigmoid monotone -> sort logits).
  #pragma unroll 1
  for (int kk = 2; kk <= KPAD; kk <<= 1) {
    #pragma unroll 1
    for (int jj = kk >> 1; jj > 0; jj >>= 1) {
      __syncthreads();
      int ixj = t ^ jj;
      if (ixj > t) {
        float a = ls[t], b = ls[ixj];
        bool d = ((t & kk) == 0);
        if ((a < b) == d) {
          ls[t] = b; ls[ixj] = a;
          int tmp = li[t]; li[t] = li[ixj]; li[ixj] = tmp;
        }
      }
    }
  }
  __syncthreads();

  // Decode + clip + store.
  {
    int i = li[t];
    float score = NEG_INF_F;
    float b0 = 0.f, b1v = 0.f, b2v = 0.f, b3 = 0.f;
    if (i >= 0 && t < KSEL) {
      score = 1.0f / (1.0f + expf(-ls[t]));
      int a = i % NA, hw = i / NA;
      float d0 = brgn[(a * 4 + 0) * HW + hw], d1 = brgn[(a * 4 + 1) * HW + hw];
      float d2 = brgn[(a * 4 + 2) * HW + hw], d3 = brgn[(a * 4 + 3) * HW + hw];
      const float* ap = ancn + (size_t)i * 4;
      float x1 = ap[0], y1 = ap[1], x2 = ap[2], y2 = ap[3];
      float w = x2 - x1 + 1.0f, h = y2 - y1 + 1.0f;
      float cx = x1 + 0.5f * w, cy = y1 + 0.5f * h;
      d2 = fminf(d2, XCLIP); d3 = fminf(d3, XCLIP);
      float pcx = d0 * w + cx, pcy = d1 * h + cy;
      float pw = expf(d2) * w, ph = expf(d3) * h;
      b0 = pcx - 0.5f * pw; b1v = pcy - 0.5f * ph;
      b2v = pcx + 0.5f * pw - 1.0f; b3 = pcy + 0.5f * ph - 1.0f;
      b0  = fminf(fmaxf(b0, 0.0f), IMW - 1.0f);
      b1v = fminf(fmaxf(b1v, 0.0f), IMH - 1.0f);
      b2v = fminf(fmaxf(b2v, 0.0f), IMW - 1.0f);
      b3  = fminf(fmaxf(b3, 0.0f), IMH - 1.0f);
    }
    size_t o = (size_t)blk * KPAD + t;
    sel_score[o] = score;
    sel_idx[o] = i;
    float* bp = sel_box + o * 4;
    bp[0] = b0; bp[1] = b1v; bp[2] = b2v; bp[3] = b3;
  }
}

// ---------------------------------------------------------------------------
// Kernel 2: greedy NMS per (img,level), two-phase bitmask formulation.
//  Phase A: 1000 threads each build their 1024-bit suppression row in LDS
//           (125 KB matrix -- only possible with CDNA5's 320 KB WGP LDS).
//           Unrolling bounded to keep the kernel inside the I-cache.
//  Phase B: one wave32 holds the keep mask in registers (1 word/lane) and
//           replays the greedy scan with zero block barriers.
// Boxes are staged via CDNA5 async global->LDS DMA.
// Dynamic LDS: 148128 bytes, requested at launch.
// ---------------------------------------------------------------------------
extern "C" __global__ __launch_bounds__(1024)
void k_nms(const float* __restrict__ sel_score, const float* __restrict__ sel_box,
           float* __restrict__ msk_score)
{
  extern __shared__ char smem[];
  float*    bx    = (float*)smem;               // KSEL*4 floats  (16000 B)
  float*    ar    = bx + KSEL * 4;              // KSEL floats    ( 4000 B)
  unsigned* sup   = (unsigned*)(ar + KSEL);     // KSEL*32 words  (128000 B)
  unsigned* keepw = sup + KSEL * 32;            // 32 words       (  128 B)

  const int blk = blockIdx.x, t = threadIdx.x;
  const float* src = sel_box + (size_t)blk * KPAD * 4;

#if USE_ASYNC_LDS
  typedef int v4i_t __attribute__((__vector_size__(16)));
  typedef __attribute__((address_space(1))) v4i_t gv4i_t;
  typedef __attribute__((address_space(3))) v4i_t lv4i_t;
  if (t < KSEL) {
    gv4i_t* g = (gv4i_t*)(unsigned long long)(const void*)(src + (size_t)t * 4);
    lv4i_t* l = (lv4i_t*)(unsigned)(size_t)(void*)(&bx[t * 4]);
    __builtin_amdgcn_global_load_async_to_lds_b128(g, l, 0, 0);
  }
  __builtin_amdgcn_s_wait_asynccnt(0);
  __syncthreads();
#else
  if (t < KSEL) {
    const float4 v = ((const float4*)src)[t];
    bx[t * 4 + 0] = v.x; bx[t * 4 + 1] = v.y; bx[t * 4 + 2] = v.z; bx[t * 4 + 3] = v.w;
  }
  __syncthreads();
#endif

  if (t < KSEL) {
    float x0 = bx[t * 4 + 0], y0 = bx[t * 4 + 1], x1 = bx[t * 4 + 2], y1 = bx[t * 4 + 3];
    ar[t] = (x1 - x0 + 1.0f) * (y1 - y0 + 1.0f);
  }
  __syncthreads();

  // Phase A: suppression rows. Row t, bit j set iff j > t and IoU(t,j) > 0.7.
  // Division-free test: inter/(ai+aj-inter) > th  <=>  inter*(1+th) > th*(ai+aj).
  if (t < KSEL) {
    const float x0 = bx[t * 4 + 0], y0 = bx[t * 4 + 1];
    const float x1 = bx[t * 4 + 2], y1 = bx[t * 4 + 3];
    const float at = ar[t];
    const int w0 = t >> 5;
    #pragma unroll 1
    for (int w = 0; w < 32; ++w) {
      unsigned m = 0u;
      if (w >= w0) {
        #pragma unroll 4
        for (int b = 0; b < 32; ++b) {
          const int j = w * 32 + b;
          if (j > t && j < KSEL) {
            float xx1 = fmaxf(x0, bx[j * 4 + 0]);
            float yy1 = fmaxf(y0, bx[j * 4 + 1]);
            float xx2 = fminf(x1, bx[j * 4 + 2]);
            float yy2 = fminf(y1, bx[j * 4 + 3]);
            float iw = fmaxf(xx2 - xx1 + 1.0f, 0.0f);
            float ih = fmaxf(yy2 - yy1 + 1.0f, 0.0f);
            float inter = iw * ih;
            if (inter * (1.0f + NMS_TH) > NMS_TH * (at + ar[j])) m |= (1u << b);
          }
        }
      }
      sup[t * 32 + w] = m;
    }
  }
  __syncthreads();

  // Phase B: greedy scan on a single wave32 (threads 0..31 == wave 0).
  // Lane w owns keep bits for boxes [w*32, w*32+31]; no block barriers needed.
  if (t < 32) {
    const int base = t * 32;
    unsigned keep = (base + 32 <= KSEL) ? 0xFFFFFFFFu
                  : ((base >= KSEL) ? 0u : ((1u << (KSEL - base)) - 1u));
    #pragma unroll 1
    for (int i = 0; i < KSEL; ++i) {
      unsigned wi = __shfl(keep, i >> 5, 32);   // broadcast word holding bit i
      if ((wi >> (i & 31)) & 1u)                // box i survives -> apply its row
        keep &= ~sup[i * 32 + t];
    }
    keepw[t] = keep;
  }
  __syncthreads();

  float s = NEG_INF_F;
  if (t < KSEL && ((keepw[t >> 5] >> (t & 31)) & 1u))
    s = sel_score[(size_t)blk * KPAD + t];
  msk_score[(size_t)blk * KPAD + t] = s;
}

#define NMS_SMEM_BYTES ((KSEL * 4 + KSEL) * (int)sizeof(float) + (KSEL * 32 + 32) * (int)sizeof(unsigned))

// ---------------------------------------------------------------------------
// Kernel 3: per image, sort the 3x1000 masked candidates (bitonic over 4096),
// emit top-1000 boxes/scores and FPN level assignment.
// ---------------------------------------------------------------------------
extern "C" __global__ __launch_bounds__(1024)
void k_final(const float* __restrict__ msk_score, const float* __restrict__ sel_box,
             float* __restrict__ out_box, float* __restrict__ out_score, int* __restrict__ out_lvl)
{
  const int img = blockIdx.x, t = threadIdx.x, NT = blockDim.x;
  __shared__ float s[FINPAD];
  __shared__ int   id[FINPAD];
  for (int j = t; j < FINPAD; j += NT) { s[j] = NEG_INF_F; id[j] = -1; }
  __syncthreads();
  for (int lvl = 0; lvl < NLVL; lvl++) {
    const float* m = msk_score + (size_t)(img * NLVL + lvl) * KPAD;
    for (int j = t; j < KSEL; j += NT) {
      s[lvl * KSEL + j] = m[j];
      id[lvl * KSEL + j] = lvl * KPAD + j;
    }
  }
  __syncthreads();

  #pragma unroll 1
  for (int kk = 2; kk <= FINPAD; kk <<= 1) {
    #pragma unroll 1
    for (int jj = kk >> 1; jj > 0; jj >>= 1) {
      #pragma unroll 1
      for (int m2 = t; m2 < (FINPAD / 2); m2 += NT) {
        int idx = ((m2 & ~(jj - 1)) << 1) | (m2 & (jj - 1));
        int p = idx | jj;
        bool d = ((idx & kk) == 0);
        float a = s[idx], b = s[p];
        if ((a < b) == d) {
          s[idx] = b; s[p] = a;
          int tmp = id[idx]; id[idx] = id[p]; id[p] = tmp;
        }
      }
      __syncthreads();
    }
  }

  if (t < KSEL) {
    float scv = s[t];
    int sid = id[t];
    float b0 = 0.f, b1 = 0.f, b2 = 0.f, b3 = 0.f;
    if (sid >= 0) {
      const float* bp = sel_box + ((size_t)img * NLVL * KPAD + (size_t)sid) * 4;
      b0 = bp[0]; b1 = bp[1]; b2 = bp[2]; b3 = bp[3];
    }
    size_t o = (size_t)img * KSEL + t;
    float* ob = out_box + o * 4;
    ob[0] = b0; ob[1] = b1; ob[2] = b2; ob[3] = b3;
    out_score[o] = scv;
    float area = (b2 - b0 + 1.0f) * (b3 - b1 + 1.0f);
    float lf = floorf(4.0f + log2f(sqrtf(area) / 224.0f + 1e-6f));
    int lv = (int)lf;
    lv = lv < 2 ? 2 : (lv > 5 ? 5 : lv);
    out_lvl[o] = lv;
  }
}

// ---------------------------------------------------------------------------
extern "C" void kernel_launch(void* const* d_in, const int* in_sizes, int n_in,
                              void* d_out, int out_size, void* d_ws, size_t ws_size,
                              hipStream_t stream)
{
  const float* obj0 = (const float*)d_in[0];
  const float* obj1 = (const float*)d_in[1];
  const float* obj2 = (const float*)d_in[2];
  const float* brg0 = (const float*)d_in[3];
  const float* brg1 = (const float*)d_in[4];
  const float* brg2 = (const float*)d_in[5];
  const float* anc0 = (const float*)d_in[6];
  const float* anc1 = (const float*)d_in[7];
  const float* anc2 = (const float*)d_in[8];

  const size_t nsel = (size_t)NIMG * NLVL * KPAD;
  float* sel_score = (float*)d_ws;
  float* sel_box   = sel_score + nsel;        // nsel*4 floats
  int*   sel_idx   = (int*)(sel_box + nsel * 4);
  float* msk       = (float*)(sel_idx + nsel);

  k_topk_decode<<<NIMG * NLVL, 1024, 0, stream>>>(obj0, obj1, obj2, brg0, brg1, brg2,
                                                  anc0, anc1, anc2,
                                                  sel_score, sel_box, sel_idx);
  k_nms<<<NIMG * NLVL, 1024, NMS_SMEM_BYTES, stream>>>(sel_score, sel_box, msk);

  float* out_box   = (float*)d_out;                                   // 8*1000*4
  float* out_score = out_box + (size_t)NIMG * KSEL * 4;               // 8*1000
  int*   out_lvl   = (int*)d_out + (size_t)NIMG * KSEL * 4 + (size_t)NIMG * KSEL;
  k_final<<<NIMG, 1024, 0, stream>>>(msk, sel_box, out_box, out_score, out_lvl);

  (void)in_sizes; (void)n_in; (void)out_size; (void)ws_size;
}